// CapsLayerQuat_44023414784333
// MI455X (gfx1250) — compile-verified
//
#include <hip/hip_runtime.h>
#include <math.h>

typedef float v2f __attribute__((ext_vector_type(2)));
typedef float v8f __attribute__((ext_vector_type(8)));
typedef unsigned int u32x4 __attribute__((ext_vector_type(4)));
typedef int i32x4 __attribute__((ext_vector_type(4)));
typedef int i32x8 __attribute__((ext_vector_type(8)));

#define EPSF      1e-8f
#define I_CAPS    16
#define O_CAPS    16
#define R_ITERS   3
#define SPB       16          // samples per block (one 16-row WMMA tile)
#define BLOCK_THREADS 128     // 4 waves (wave32)
#define MT_STRIDE 65          // padded row stride for B-matrix table (bank spread)
#define XS_STRIDE 68          // padded per-sample stride for staged x (64 + 4 pad)

// LDS layout (dynamic shared):
//   mt   [4*I][MT_STRIDE] : B columns, mt[(i*4+e)*MT_STRIDE + (o*4+d)] = M[o,i][d][e]
//   trc  [I][64]          : C columns, trc[i*64 + o*4+d] = (d? trans[o,i,d-1] : 0)
//   xs   [SPB][XS_STRIDE] : staged x tile (TDM-written, padded rows)
//   votes[SPB][I][O][4]   : WMMA output, consumed by routing
#define MT_FLOATS   (4 * I_CAPS * MT_STRIDE)
#define TRC_FLOATS  (I_CAPS * 64)
#define XS_FLOATS   (SPB * XS_STRIDE)
#define VOTE_FLOATS (SPB * I_CAPS * O_CAPS * 4)
#define SMEM_FLOATS (MT_FLOATS + TRC_FLOATS + XS_FLOATS + VOTE_FLOATS)
#define XS_BYTE_OFF ((MT_FLOATS + TRC_FLOATS) * 4)

#if defined(__has_builtin)
#if __has_builtin(__builtin_amdgcn_tensor_load_to_lds) && \
    __has_builtin(__builtin_amdgcn_s_wait_tensorcnt)
#define USE_TDM 1
#endif
#endif

__global__ __launch_bounds__(BLOCK_THREADS)
void caps_quat_wmma_kernel(const float* __restrict__ x,
                           const float* __restrict__ quats,
                           const float* __restrict__ scale,
                           const float* __restrict__ trans,
                           const float* __restrict__ bias,
                           const float* __restrict__ beta,
                           const float* __restrict__ alpha,
                           float* __restrict__ out) {
  extern __shared__ float smem[];
  float* mt    = smem;
  float* trc   = mt  + MT_FLOATS;
  float* xs    = trc + TRC_FLOATS;
  float* votes = xs  + XS_FLOATS;

  const int tid  = threadIdx.x;
  const int lane = tid & 31;
  const int wave = tid >> 5;
  const int n0   = blockIdx.x * SPB;

#ifdef USE_TDM
  // ---- Kick off TDM: DMA x tile (16 samples x 64 f32) HBM -> LDS, with LDS
  // padding of 4 DWORDs every 64 DWORDs so rows land at XS_STRIDE=68.
  // D# per ISA ch.8: group0 = {count|..., lds_addr, global_addr lo/hi|type=2},
  // group1 = {data_size=4B|pad cfg, dims/strides}, groups 2/3 zero (2D tensor).
  if (wave == 0) {
    const unsigned long long gaddr =
        (unsigned long long)(const void*)(x + (size_t)n0 * (I_CAPS * 4));
    u32x4 g0;
    g0.x = 1u;                                        // count=1, user mode
    g0.y = (unsigned int)XS_BYTE_OFF;                 // lds_addr (bytes)
    g0.z = (unsigned int)(gaddr & 0xFFFFFFFFu);       // global_addr[31:0]
    g0.w = (unsigned int)((gaddr >> 32) & 0x01FFFFFFu) | (2u << 30); // [56:32]|type=2
    i32x8 g1;
    g1[0] = (2 << 16)      // data_size = 4 bytes
          | (1 << 20)      // pad_enable
          | (5 << 22)      // pad_interval: every 64 DWORDs
          | (3 << 25);     // pad_amount: 4 DWORDs
    g1[1] = (64 << 16);    // tensor_dim0[15:0] = 64 elements/row
    g1[2] = (SPB << 16);   // tensor_dim1[15:0] = 16 rows
    g1[3] = (64 << 16);    // tile_dim0 = 64
    g1[4] = SPB;           // tile_dim1 = 16, tile_dim2 = 0 (2D)
    g1[5] = 64;            // tensor_dim0_stride[31:0] = 64
    g1[6] = 0; g1[7] = 0;
    i32x4 gz = {0, 0, 0, 0};
#if defined(__clang_major__) && (__clang_major__ >= 23)
    i32x8 gz8 = {0, 0, 0, 0, 0, 0, 0, 0};
    __builtin_amdgcn_tensor_load_to_lds(g0, g1, gz, gz, gz8, 0);  // 6-arg form
#else
    __builtin_amdgcn_tensor_load_to_lds(g0, g1, gz, gz, 0);       // 5-arg form
#endif
  }
#endif

  // ---------------- Phase 0: rotation matrices + trans columns (overlaps TDM)
  // votes_d = sum_e M[d][e] * x_e, with qn = q * rsqrt(|q|^2+eps), M scaled by `scale`.
  for (int p = tid; p < O_CAPS * I_CAPS; p += BLOCK_THREADS) {
    const int oo = p >> 4, ii = p & 15;
    float4 q = *(const float4*)&quats[p * 4];
    float rn = scale[p] * rsqrtf(q.x*q.x + q.y*q.y + q.z*q.z + q.w*q.w + EPSF);
    float w = q.x * rn, xq = q.y * rn, yq = q.z * rn, zq = q.w * rn;
    const int cb = oo * 4;                       // column base (o*4 + d)
    float* m0 = &mt[(ii*4 + 0) * MT_STRIDE + cb];   // e=0 coeff for d=0..3
    m0[0] =  w;  m0[1] =  xq; m0[2] =  yq; m0[3] =  zq;
    float* m1 = &mt[(ii*4 + 1) * MT_STRIDE + cb];   // e=1
    m1[0] = -xq; m1[1] =  w;  m1[2] =  zq; m1[3] = -yq;
    float* m2 = &mt[(ii*4 + 2) * MT_STRIDE + cb];   // e=2
    m2[0] = -yq; m2[1] = -zq; m2[2] =  w;  m2[3] =  xq;
    float* m3 = &mt[(ii*4 + 3) * MT_STRIDE + cb];   // e=3
    m3[0] = -zq; m3[1] =  yq; m3[2] = -xq; m3[3] =  w;
    trc[ii*64 + cb + 0] = 0.f;
    trc[ii*64 + cb + 1] = trans[p*3 + 0];
    trc[ii*64 + cb + 2] = trans[p*3 + 1];
    trc[ii*64 + cb + 3] = trans[p*3 + 2];
  }
#ifdef USE_TDM
  if (wave == 0) __builtin_amdgcn_s_wait_tensorcnt(0);
#else
  {
    const float* xblk = x + (size_t)n0 * (I_CAPS * 4);
    for (int idx = tid; idx < SPB * I_CAPS * 4; idx += BLOCK_THREADS)
      xs[(idx >> 6) * XS_STRIDE + (idx & 63)] = xblk[idx];
  }
#endif
  __syncthreads();

  // ---------------- Phase 1: votes via V_WMMA_F32_16X16X4_F32 ---------------
  // Per i: D[16 samples x 16 (o,d)-cols] = A[16x4 x-rows] * B[4x16 M-cols] + C(trans).
  // A layout: lanes 0-15 rows, v0/v1 = K0/K1; lanes 16-31 same rows, K2/K3.
  {
    const int mrow = lane & 15;
    const int kb   = (lane >> 4) * 2;
    const int half = (lane >> 4) * 8;
#pragma unroll
    for (int ic = 0; ic < 4; ++ic) {                  // 4 i's per wave, 16 WMMAs
      const int i = wave * 4 + ic;
      v2f afrag;
      afrag.x = xs[mrow * XS_STRIDE + i * 4 + kb];
      afrag.y = xs[mrow * XS_STRIDE + i * 4 + kb + 1];
#pragma unroll
      for (int c = 0; c < 4; ++c) {
        const int col = c * 16 + mrow;                // global column = o*4 + d
        v2f bfrag;
        bfrag.x = mt[(i * 4 + kb)     * MT_STRIDE + col];
        bfrag.y = mt[(i * 4 + kb + 1) * MT_STRIDE + col];
        const float cv = trc[i * 64 + col];
        v8f acc = {cv, cv, cv, cv, cv, cv, cv, cv};   // trans folded into C
        v8f dm = __builtin_amdgcn_wmma_f32_16x16x4_f32(
            false, afrag, false, bfrag, (short)0, acc, false, false);
        const int og = col >> 2, dg = col & 3;
#pragma unroll
        for (int j = 0; j < 8; ++j) {                 // D: vgpr j -> row j (+8 hi half)
          const int nl = j + half;
          votes[((nl * I_CAPS + i) * O_CAPS + og) * 4 + dg] = dm[j];
        }
      }
    }
  }
  __syncthreads();

  // ---------------- Phase 2: dynamic routing, one wave per sample -----------
  // lane -> (o = lane>>1, i = (lane&1)*8 + j, j<8); 32 vote floats in VGPRs.
  {
    const int o     = lane >> 1;
    const int ibase = (lane & 1) * 8;
    const float pbeta = beta[o], pbab = alpha[o] + bias[o];

    for (int k = 0; k < SPB / 4; ++k) {
      const int nl = wave * 4 + k;
      float4 vt[8];
#pragma unroll
      for (int j = 0; j < 8; ++j)
        vt[j] = *(const float4*)&votes[((nl * I_CAPS + (ibase + j)) * O_CAPS + o) * 4];

      float b[8];
#pragma unroll
      for (int j = 0; j < 8; ++j) b[j] = 0.f;

      float4 v = {0.f, 0.f, 0.f, 0.f};
      float n2 = 0.f;

      for (int it = 0; it < R_ITERS; ++it) {
        // softmax over o (16 lanes at stride 2 share an i): xor-reduce 2,4,8,16
        float mx[8], ex[8], sm[8];
#pragma unroll
        for (int j = 0; j < 8; ++j) mx[j] = b[j];
#pragma unroll
        for (int msk = 2; msk < 32; msk <<= 1)
#pragma unroll
          for (int j = 0; j < 8; ++j) mx[j] = fmaxf(mx[j], __shfl_xor(mx[j], msk, 32));
#pragma unroll
        for (int j = 0; j < 8; ++j) { ex[j] = __expf(b[j] - mx[j]); sm[j] = ex[j]; }
#pragma unroll
        for (int msk = 2; msk < 32; msk <<= 1)
#pragma unroll
          for (int j = 0; j < 8; ++j) sm[j] += __shfl_xor(sm[j], msk, 32);

        // s[o,:] = sum_i c * votes  (8 local i's, then pair-sum over the other 8)
        float4 s = {0.f, 0.f, 0.f, 0.f};
#pragma unroll
        for (int j = 0; j < 8; ++j) {
          const float c = ex[j] / sm[j];
          s.x = fmaf(c, vt[j].x, s.x); s.y = fmaf(c, vt[j].y, s.y);
          s.z = fmaf(c, vt[j].z, s.z); s.w = fmaf(c, vt[j].w, s.w);
        }
        s.x += __shfl_xor(s.x, 1, 32); s.y += __shfl_xor(s.y, 1, 32);
        s.z += __shfl_xor(s.z, 1, 32); s.w += __shfl_xor(s.w, 1, 32);

        n2 = s.x*s.x + s.y*s.y + s.z*s.z + s.w*s.w;
        const float coef = (n2 / (1.f + n2)) * rsqrtf(n2 + EPSF);   // squash
        v.x = s.x * coef; v.y = s.y * coef; v.z = s.z * coef; v.w = s.w * coef;

        // b[o,i] += v . votes[o,i,:]
#pragma unroll
        for (int j = 0; j < 8; ++j)
          b[j] += v.x*vt[j].x + v.y*vt[j].y + v.z*vt[j].z + v.w*vt[j].w;
      }

      const float norm_s = sqrtf(n2 + EPSF);
      const float t = fmaf(pbeta, norm_s, pbab);
      const float a = 1.f / (1.f + __expf(-t));
      if (!(lane & 1)) {                              // 16 lanes write 16 o's
        float4 oq = {v.x * a, v.y * a, v.z * a, v.w * a};
        *(float4*)&out[((size_t)(n0 + nl) * O_CAPS + o) * 4] = oq;
      }
    }
  }
}

extern "C" void kernel_launch(void* const* d_in, const int* in_sizes, int n_in,
                              void* d_out, int out_size, void* d_ws, size_t ws_size,
                              hipStream_t stream) {
  (void)n_in; (void)out_size; (void)d_ws; (void)ws_size;
  const float* x     = (const float*)d_in[0];
  const float* quats = (const float*)d_in[1];
  const float* scale = (const float*)d_in[2];
  const float* trans = (const float*)d_in[3];
  const float* bias  = (const float*)d_in[4];
  const float* beta  = (const float*)d_in[5];
  const float* alpha = (const float*)d_in[6];
  float* out = (float*)d_out;

  const int N = in_sizes[0] / (I_CAPS * 4);
  const int blocks = N / SPB;
  const size_t shmem = SMEM_FLOATS * sizeof(float);
  caps_quat_wmma_kernel<<<blocks, BLOCK_THREADS, shmem, stream>>>(
      x, quats, scale, trans, bias, beta, alpha, out);
}